// S4Layer_43602507989838
// MI455X (gfx1250) — compile-verified
//
#include <hip/hip_runtime.h>

// ---------------------------------------------------------------------------
// S4 layer for MI455X (gfx1250, wave32). All matrix math via
// V_WMMA_F32_16X16X4_F32 (full fp32 precision; the 2047-step recurrence
// makes low-precision inputs risky). Scan is chunked: depth 2047 -> ~160.
// GEMM kernels use 32-row M-tiles (2 stacked WMMA A-tiles) so each B load
// feeds 2 WMMAs -> halves L2 traffic vs 16-row tiles.
// ---------------------------------------------------------------------------

typedef __attribute__((ext_vector_type(2))) float v2f;
typedef __attribute__((ext_vector_type(8))) float v8f;

#define DMODEL   1024
#define LSEQ     2048
#define BATCH    8
#define TSTEPS   (LSEQ - 1)        // 2047 recurrence steps
#define CHUNK    64
#define NCHUNK   32                // ceil(2047/64)
#define STATE_N  (BATCH * DMODEL)  // 8192 floats per state
#define LDSPITCH 1028              // 1024 + 4 pad -> conflict-free 16-row reads

// D(16x16) = A(16x4) x B(4x16) + C, fp32 in/out.
// A: lane(l<16)=row M, vgpr v = K = 2*(l>=16)+v.  B mirrored with N=lane.
__device__ __forceinline__ v8f wmma4(v2f a, v2f b, v8f c) {
  return __builtin_amdgcn_wmma_f32_16x16x4_f32(
      /*neg_a=*/false, a, /*neg_b=*/false, b,
      /*c_mod=*/(short)0, c, /*reuse_a=*/false, /*reuse_b=*/false);
}

// ---------------------------------------------------------------------------
// WxT[d][k] = rc_w[k][d];  WyT[d][k] = rc_w[k][D+d] + rc_w[k][2D+d]
// ---------------------------------------------------------------------------
__global__ void k_build_wt(const float* __restrict__ rc_w,
                           float* __restrict__ WxT, float* __restrict__ WyT) {
  int k = blockIdx.x;
  int d = blockIdx.y * 256 + threadIdx.x;
  long base = (long)k * (3 * DMODEL);
  float wx = rc_w[base + d];
  float wy = rc_w[base + DMODEL + d] + rc_w[base + 2 * DMODEL + d];
  WxT[(long)d * DMODEL + k] = wx;
  WyT[(long)d * DMODEL + k] = wy;
}

// ---------------------------------------------------------------------------
// Generic 1024x1024x1024 GEMM:  C[i][j] = sum_k A[i][k] * B[j][k]  (A @ B^T)
// Block = 32-row M-tile (two stacked WMMA tiles) x full N (1024).
// 32 waves x 2 N-tiles; each B load pair feeds 2 WMMAs. Grid = 32.
// ---------------------------------------------------------------------------
__global__ __launch_bounds__(1024)
void k_gemm_nt(const float* __restrict__ A, const float* __restrict__ B,
               float* __restrict__ C) {
  __shared__ float As[32 * LDSPITCH];
  int tid = threadIdx.x;
  int tile0 = blockIdx.x * 32;
  for (int idx = tid; idx < 32 * DMODEL; idx += 1024) {
    int m = idx >> 10, k = idx & 1023;
    As[m * LDSPITCH + k] = A[(long)(tile0 + m) * DMODEL + k];
  }
  __syncthreads();
  int wave = tid >> 5, lane = tid & 31;
  int ml = lane & 15, hi = lane >> 4;
  int n0a = wave * 32, n0b = n0a + 16;
  const float* ba = B + (long)(n0a + ml) * DMODEL;
  const float* bb = B + (long)(n0b + ml) * DMODEL;
  const float* arLo = &As[ml * LDSPITCH];
  const float* arHi = &As[(16 + ml) * LDSPITCH];
  v8f acc00 = {}, acc01 = {}, acc10 = {}, acc11 = {};
#pragma unroll 4
  for (int k0 = 0; k0 < DMODEL; k0 += 4) {
    int k = k0 + 2 * hi;
    v2f a0 = *(const v2f*)(arLo + k);
    v2f a1 = *(const v2f*)(arHi + k);
    v2f b0 = *(const v2f*)(ba + k);
    v2f b1 = *(const v2f*)(bb + k);
    acc00 = wmma4(a0, b0, acc00);
    acc01 = wmma4(a0, b1, acc01);
    acc10 = wmma4(a1, b0, acc10);
    acc11 = wmma4(a1, b1, acc11);
  }
#pragma unroll
  for (int r = 0; r < 8; ++r) {
    int m = r + 8 * hi;                       // D layout: vgpr r, hi half -> M=r+8
    long rowLo = (long)(tile0 + m) * DMODEL;
    long rowHi = (long)(tile0 + 16 + m) * DMODEL;
    C[rowLo + n0a + ml] = acc00[r];
    C[rowLo + n0b + ml] = acc01[r];
    C[rowHi + n0a + ml] = acc10[r];
    C[rowHi + n0b + ml] = acc11[r];
  }
}

// ---------------------------------------------------------------------------
// cvec[e] = sum_k M[e][k] * rc_b[k]
// ---------------------------------------------------------------------------
__global__ void k_matvec(const float* __restrict__ Mw,
                         const float* __restrict__ b, float* __restrict__ cvec) {
  __shared__ float red[256];
  int e = blockIdx.x;
  float s = 0.f;
  for (int k = threadIdx.x; k < DMODEL; k += 256)
    s += Mw[(long)e * DMODEL + k] * b[k];
  red[threadIdx.x] = s;
  __syncthreads();
  for (int off = 128; off > 0; off >>= 1) {
    if (threadIdx.x < off) red[threadIdx.x] += red[threadIdx.x + off];
    __syncthreads();
  }
  if (threadIdx.x == 0) cvec[e] = red[0];
}

// ---------------------------------------------------------------------------
// u[t][bb][e] = sum_d x[bb][t+1][d] * P[e][d] + c[e]
// Rows r = bb*2047 + t tiled by 32 (two stacked WMMA tiles). Grid = 512.
// ---------------------------------------------------------------------------
__global__ __launch_bounds__(1024)
void k_u(const float* __restrict__ x, const float* __restrict__ P,
         const float* __restrict__ cvec, float* __restrict__ U) {
  __shared__ float As[32 * LDSPITCH];
  const int MR = BATCH * TSTEPS;  // 16376
  int tid = threadIdx.x;
  int tile0 = blockIdx.x * 32;
  for (int idx = tid; idx < 32 * DMODEL; idx += 1024) {
    int m = idx >> 10, k = idx & 1023;
    int R = tile0 + m;
    float v = 0.f;
    if (R < MR) {
      int bb = R / TSTEPS, t = R - bb * TSTEPS;
      v = x[((long)bb * LSEQ + (t + 1)) * DMODEL + k];
    }
    As[m * LDSPITCH + k] = v;
  }
  __syncthreads();
  int wave = tid >> 5, lane = tid & 31;
  int ml = lane & 15, hi = lane >> 4;
  int n0a = wave * 32, n0b = n0a + 16;
  const float* ba  = P + (long)(n0a + ml) * DMODEL;
  const float* bb_ = P + (long)(n0b + ml) * DMODEL;
  const float* arLo = &As[ml * LDSPITCH];
  const float* arHi = &As[(16 + ml) * LDSPITCH];
  v8f acc00 = {}, acc01 = {}, acc10 = {}, acc11 = {};
#pragma unroll 4
  for (int k0 = 0; k0 < DMODEL; k0 += 4) {
    int k = k0 + 2 * hi;
    v2f a0 = *(const v2f*)(arLo + k);
    v2f a1 = *(const v2f*)(arHi + k);
    v2f b0 = *(const v2f*)(ba + k);
    v2f b1 = *(const v2f*)(bb_ + k);
    acc00 = wmma4(a0, b0, acc00);
    acc01 = wmma4(a0, b1, acc01);
    acc10 = wmma4(a1, b0, acc10);
    acc11 = wmma4(a1, b1, acc11);
  }
  float c0 = cvec[n0a + ml], c1 = cvec[n0b + ml];
#pragma unroll
  for (int r = 0; r < 8; ++r) {
    int m = r + 8 * hi;
#pragma unroll
    for (int h = 0; h < 2; ++h) {
      int R = tile0 + 16 * h + m;
      if (R < MR) {
        int bbi = R / TSTEPS, t = R - bbi * TSTEPS;
        long row = ((long)t * BATCH + bbi) * DMODEL;
        U[row + n0a + ml] = (h ? acc10[r] : acc00[r]) + c0;
        U[row + n0b + ml] = (h ? acc11[r] : acc01[r]) + c1;
      }
    }
  }
}

// ---------------------------------------------------------------------------
// Chunk scan: y_t = u_t + Qm * y_{t-1}, state 8x1024 (padded to 16) in LDS
// ping-pong buffers. One workgroup per chunk; only workgroup barriers.
//   init   : per-chunk initial state, stride init_cs (floats)
//   U      : u sequence, [t][8][1024], block base = c*chunkT*8192
//   outSeq : optional, addr = base + c*chunkT*ostT + t*ostT + m*ostM + n
//   outFinal: optional final state per chunk, [c][8][1024]
// ---------------------------------------------------------------------------
__global__ __launch_bounds__(1024)
void k_scan(const float* __restrict__ init, long init_cs,
            const float* __restrict__ U, const float* __restrict__ Qm,
            float* __restrict__ outSeq, long ostT, long ostM,
            float* __restrict__ outFinal, int totalT, int chunkT) {
  __shared__ float ybuf[2 * 16 * LDSPITCH];
  int tid = threadIdx.x;
  int c = blockIdx.x;
  int Tc = totalT - c * chunkT;
  if (Tc > chunkT) Tc = chunkT;
  if (Tc <= 0) return;

  for (int idx = tid; idx < 16 * DMODEL; idx += 1024) {
    int m = idx >> 10, k = idx & 1023;
    float v = (m < BATCH) ? init[(long)c * init_cs + m * DMODEL + k] : 0.f;
    ybuf[m * LDSPITCH + k] = v;                      // buf0: state, pad=0
    ybuf[16 * LDSPITCH + m * LDSPITCH + k] = 0.f;    // buf1: zero (pad stays 0)
  }
  __syncthreads();

  int wave = tid >> 5, lane = tid & 31;
  int ml = lane & 15, hi = lane >> 4;
  int n0a = wave * 32, n0b = n0a + 16;
  const float* qa = Qm + (long)(n0a + ml) * DMODEL;
  const float* qb = Qm + (long)(n0b + ml) * DMODEL;
  const float* Ublk = U + (long)c * chunkT * STATE_N;
  float* outBlk = outSeq ? (outSeq + (long)c * chunkT * ostT) : (float*)0;

  for (int t = 0; t < Tc; ++t) {
    const float* yr = &ybuf[(t & 1) * 16 * LDSPITCH + ml * LDSPITCH];
    v8f acc0 = {}, acc1 = {};
#pragma unroll 4
    for (int k0 = 0; k0 < DMODEL; k0 += 4) {
      int k = k0 + 2 * hi;
      v2f a  = *(const v2f*)(yr + k);
      v2f b0 = *(const v2f*)(qa + k);
      v2f b1 = *(const v2f*)(qb + k);
      acc0 = wmma4(a, b0, acc0);
      acc1 = wmma4(a, b1, acc1);
    }
    float* yw = &ybuf[((t & 1) ^ 1) * 16 * LDSPITCH];
    if (hi == 0) {  // rows 0..7 are the real batch rows
      const float* ut = Ublk + (long)t * STATE_N;
#pragma unroll
      for (int r = 0; r < 8; ++r) {
        float v0 = acc0[r] + ut[r * DMODEL + n0a + ml];
        float v1 = acc1[r] + ut[r * DMODEL + n0b + ml];
        yw[r * LDSPITCH + n0a + ml] = v0;
        yw[r * LDSPITCH + n0b + ml] = v1;
        if (outBlk) {
          outBlk[(long)t * ostT + (long)r * ostM + n0a + ml] = v0;
          outBlk[(long)t * ostT + (long)r * ostM + n0b + ml] = v1;
        }
      }
    }
    __syncthreads();
  }
  if (outFinal) {
    const float* yf = &ybuf[(Tc & 1) * 16 * LDSPITCH];
    for (int idx = tid; idx < STATE_N; idx += 1024) {
      int m = idx >> 10, k = idx & 1023;
      outFinal[(long)c * STATE_N + idx] = yf[m * LDSPITCH + k];
    }
  }
}

// I[0] = y0 = x[:,0,:], I[1..31] = 0; also d_out[:,0,:] = y0
__global__ void k_init(const float* __restrict__ x, float* __restrict__ I,
                       float* __restrict__ out) {
  int idx = blockIdx.x * 256 + threadIdx.x;  // < 32*8192
  int c = idx >> 13;
  int m = (idx >> 10) & 7;
  int n = idx & 1023;
  float v = 0.f;
  if (c == 0) {
    v = x[(long)m * LSEQ * DMODEL + n];
    out[(long)m * LSEQ * DMODEL + n] = v;
  }
  I[idx] = v;
}

__global__ void k_copy8k(const float* __restrict__ src, float* __restrict__ dst) {
  int i = blockIdx.x * 256 + threadIdx.x;
  dst[i] = src[i];
}

// ---------------------------------------------------------------------------
extern "C" void kernel_launch(void* const* d_in, const int* in_sizes, int n_in,
                              void* d_out, int out_size, void* d_ws, size_t ws_size,
                              hipStream_t stream) {
  (void)in_sizes; (void)n_in; (void)out_size; (void)ws_size;
  const float* x    = (const float*)d_in[0];
  const float* A_w  = (const float*)d_in[1];
  const float* C_w  = (const float*)d_in[2];
  const float* rc_w = (const float*)d_in[3];
  const float* rc_b = (const float*)d_in[4];
  float* out = (float*)d_out;
  float* ws  = (float*)d_ws;

  const long MB1 = 1024L * 1024L;
  float* WxT = ws + 0 * MB1;
  float* WyT = ws + 1 * MB1;
  float* Mw  = ws + 2 * MB1;
  float* P   = ws + 3 * MB1;
  float* Q   = ws + 4 * MB1;
  float* QT  = ws + 5 * MB1;
  float* bA  = ws + 6 * MB1;
  float* bB  = ws + 7 * MB1;
  float* bC  = ws + 8 * MB1;   // Q^64 ends here
  float* bD  = ws + 9 * MB1;
  float* cv  = ws + 10 * MB1;
  float* F   = cv + 4096;                    // 32 chunk-final states
  float* I   = F + NCHUNK * STATE_N;         // 32 chunk-initial states
  float* U   = ws + 11 * MB1;                // u: [2047][8][1024] (~64 MB)

  // Weight preprocessing (all WMMA GEMMs; A_w is symmetric so NT == NN)
  k_build_wt<<<dim3(1024, 4), 256, 0, stream>>>(rc_w, WxT, WyT);
  k_gemm_nt<<<32, 1024, 0, stream>>>(C_w, A_w, Mw);   // M  = C_w @ A_w
  k_gemm_nt<<<32, 1024, 0, stream>>>(Mw, WxT, P);     // P  = M @ Wx
  k_gemm_nt<<<32, 1024, 0, stream>>>(Mw, WyT, Q);     // Q  = M @ Wy
  k_gemm_nt<<<32, 1024, 0, stream>>>(WyT, Mw, QT);    // Q^T
  k_matvec<<<1024, 256, 0, stream>>>(Mw, rc_b, cv);   // c  = M @ rc_b

  // u = x[:,1:] @ P^T + c   (big parallel GEMM, 34 GF)
  k_u<<<512, 1024, 0, stream>>>(x, P, cv, U);

  // Q^64 by repeated squaring, carrying transposes: R2 = nt(R,RT), R2T = nt(RT,R)
  k_gemm_nt<<<32, 1024, 0, stream>>>(Q,  QT, bA);     // Q^2
  k_gemm_nt<<<32, 1024, 0, stream>>>(QT, Q,  bB);     // Q^2 T
  k_gemm_nt<<<32, 1024, 0, stream>>>(bA, bB, bC);     // Q^4
  k_gemm_nt<<<32, 1024, 0, stream>>>(bB, bA, bD);     // Q^4 T
  k_gemm_nt<<<32, 1024, 0, stream>>>(bC, bD, bA);     // Q^8
  k_gemm_nt<<<32, 1024, 0, stream>>>(bD, bC, bB);     // Q^8 T
  k_gemm_nt<<<32, 1024, 0, stream>>>(bA, bB, bC);     // Q^16
  k_gemm_nt<<<32, 1024, 0, stream>>>(bB, bA, bD);     // Q^16 T
  k_gemm_nt<<<32, 1024, 0, stream>>>(bC, bD, bA);     // Q^32
  k_gemm_nt<<<32, 1024, 0, stream>>>(bD, bC, bB);     // Q^32 T
  k_gemm_nt<<<32, 1024, 0, stream>>>(bA, bB, bC);     // Q^64 -> bC

  k_init<<<1024, 256, 0, stream>>>(x, I, out);

  // Pass A: per-chunk local scans (zero init except chunk 0) -> finals F
  k_scan<<<NCHUNK, 1024, 0, stream>>>(I, (long)STATE_N, U, Q,
                                      (float*)0, 0, 0, F, TSTEPS, CHUNK);
  // Boundary states: I[1] = e_0 = F[0]; I[c] = e_{c-1} via e_c = Q^64 e_{c-1} + F_c
  k_copy8k<<<32, 256, 0, stream>>>(F, I + STATE_N);
  k_scan<<<1, 1024, 0, stream>>>(F, 0L, F + STATE_N, bC,
                                 I + 2 * STATE_N, (long)STATE_N, (long)DMODEL,
                                 (float*)0, NCHUNK - 2, NCHUNK - 2);
  // Pass C: re-scan each chunk from its true boundary state, write d_out[:,1:,:]
  k_scan<<<NCHUNK, 1024, 0, stream>>>(I, (long)STATE_N, U, Q,
                                      out + DMODEL, (long)DMODEL,
                                      (long)LSEQ * DMODEL, (float*)0,
                                      TSTEPS, CHUNK);
}